// S4DLayer_71262097376067
// MI455X (gfx1250) — compile-verified
//
#include <hip/hip_runtime.h>
#include <hip/hip_bf16.h>

// S4D bidirectional layer for gfx1250 (MI455X), chunked block-Toeplitz form.
// All heavy math runs on V_WMMA_F32_16X16X4_F32 (fp32 WMMA, exact precision).
//
// Dims (fixed by the reference): B=64, L=2048, H=256, N2=32. Chunk T=64, NC=32.
//
// ws layout:
//   WT  : v2f [2][256][32]            w^T per (dir,h,n)                (128 KB)
//   AP  : v2f [2][256][64][64]        [M_h | P_h] A-operand packed     (16 MB)
//   QP  : v2f [2][256][32][64]        Q_h A-operand packed             (8 MB)
//   WA  : v2f [2][128][512]           W_glu A-operand packed           (1 MB)
//   G   : f32 [256][B*L]              GELU activations (reused per dir)(134 MB)

typedef float v2f __attribute__((ext_vector_type(2)));
typedef float v8f __attribute__((ext_vector_type(8)));

#define NB 64
#define LL 2048
#define HH 256
#define NN 32
#define TT 64
#define NCHUNK 32
#define BL (NB * LL)

__device__ __forceinline__ v8f wmma4(v2f a, v2f b, v8f c) {
  // D = A(16x4 f32) * B(4x16 f32) + C(16x16 f32)
  return __builtin_amdgcn_wmma_f32_16x16x4_f32(
      /*neg_a=*/false, a, /*neg_b=*/false, b,
      /*c_mod=*/(short)0, c, /*reuse_a=*/false, /*reuse_b=*/false);
}

// ---------------------------------------------------------------------------
// k_prep: one block per (dir,h), 64 threads. Builds K, [M|P], Q, w^T tables.
// A-operand packing: AP[p][row] = (A[row][2p], A[row][2p+1]) so a lane loads
// one coalesced v2f: lane row = M, (k+koff, k+koff+1) per the ISA f32 A layout.
// ---------------------------------------------------------------------------
__global__ __launch_bounds__(64) void k_prep(
    const float* __restrict__ ld0,  const float* __restrict__ cr0,
    const float* __restrict__ ci0,  const float* __restrict__ lar0,
    const float* __restrict__ aim0,
    const float* __restrict__ ld1,  const float* __restrict__ cr1,
    const float* __restrict__ ci1,  const float* __restrict__ lar1,
    const float* __restrict__ aim1,
    v2f* __restrict__ WT, v2f* __restrict__ AP, v2f* __restrict__ QP) {
  const int d = blockIdx.x >> 8;
  const int h = blockIdx.x & 255;
  const float* ld  = d ? ld1  : ld0;
  const float* crp = d ? cr1  : cr0;
  const float* cip = d ? ci1  : ci0;
  const float* lar = d ? lar1 : lar0;
  const float* aim = d ? aim1 : aim0;

  __shared__ float sAr[NN], sAi[NN], sCr[NN], sCi[NN], sK[TT];
  const int tid = threadIdx.x;

  if (tid < NN) {
    const int n = tid;
    float dt  = expf(ld[h]);
    float ar0 = -expf(lar[h * NN + n]);
    float ai0 = aim[h * NN + n];
    float dr = ar0 * dt, di = ai0 * dt;           // dtA
    float e  = expf(dr);
    float wr = e * cosf(di), wi = e * sinf(di);   // w = exp(dtA)
    // (w-1)/A = (w-1)*conj(A)/|A|^2
    float den = ar0 * ar0 + ai0 * ai0;
    float xr = wr - 1.0f, xi = wi;
    float br = (xr * ar0 + xi * ai0) / den;
    float bi = (xi * ar0 - xr * ai0) / den;
    float cr = crp[h * NN + n], ci = cip[h * NN + n];
    sAr[n] = dr; sAi[n] = di;
    sCr[n] = 2.0f * (cr * br - ci * bi);          // fold the 2*Re() factor
    sCi[n] = 2.0f * (cr * bi + ci * br);
    float eT = expf(dr * (float)TT);
    v2f wT; wT.x = eT * cosf(di * (float)TT); wT.y = eT * sinf(di * (float)TT);
    WT[(size_t)(d * HH + h) * NN + n] = wT;
  }
  __syncthreads();

  {  // K[j] = Re(sum_n C2_n w_n^j)
    const int j = tid;
    float acc = 0.0f;
    for (int n = 0; n < NN; ++n) {
      float e = expf(sAr[n] * (float)j);
      float X = e * cosf(sAi[n] * (float)j);
      float Y = e * sinf(sAi[n] * (float)j);
      acc += sCr[n] * X - sCi[n] * Y;
    }
    sK[j] = acc;
  }
  __syncthreads();

  const size_t apb = (size_t)(d * HH + h) * (64 * 64);
  const int j = tid;
  // Toeplitz part: A[j][t] = K[j-t] (t<=j), k = 0..63
  for (int p = 0; p < 32; ++p) {
    int t0 = 2 * p;
    v2f v;
    v.x = (t0     <= j) ? sK[j - t0]     : 0.0f;
    v.y = (t0 + 1 <= j) ? sK[j - t0 - 1] : 0.0f;
    AP[apb + (size_t)p * TT + j] = v;
  }
  // Boundary part: rows j, k = 64+2n (coef on s_re) / 64+2n+1 (coef on s_im)
  for (int n = 0; n < NN; ++n) {
    float jp = (float)(j + 1);
    float e = expf(sAr[n] * jp);
    float X = e * cosf(sAi[n] * jp);
    float Y = e * sinf(sAi[n] * jp);
    float qr = sCr[n] * X - sCi[n] * Y;   // Re(C2 w^{j+1})
    float qi = sCr[n] * Y + sCi[n] * X;   // Im(C2 w^{j+1})
    v2f v; v.x = qr; v.y = -qi;
    AP[apb + (size_t)(32 + n) * TT + j] = v;
  }
  // Q: rows r (2n=Re, 2n+1=Im), cols t: w^{T-1-t}
  const size_t qpb = (size_t)(d * HH + h) * (32 * 64);
  const int r = tid, nn = r >> 1, isIm = r & 1;
  for (int pt = 0; pt < 32; ++pt) {
    float t1 = (float)(TT - 1 - 2 * pt), t2 = (float)(TT - 2 - 2 * pt);
    float e1 = expf(sAr[nn] * t1), e2 = expf(sAr[nn] * t2);
    float a1 = sAi[nn] * t1, a2 = sAi[nn] * t2;
    v2f v;
    v.x = isIm ? e1 * sinf(a1) : e1 * cosf(a1);
    v.y = isIm ? e2 * sinf(a2) : e2 * cosf(a2);
    QP[qpb + (size_t)pt * TT + r] = v;
  }
}

// ---------------------------------------------------------------------------
// k_packW: WA[d][p][o] = (W[o][2p], W[o][2p+1]); W is (512, 256) row-major.
// ---------------------------------------------------------------------------
__global__ __launch_bounds__(256) void k_packW(
    const float* __restrict__ W0, const float* __restrict__ W1,
    v2f* __restrict__ WA) {
  int idx = blockIdx.x * 256 + threadIdx.x;   // 2*512*128 total
  int d = idx >> 16;
  int rem = idx & 65535;
  int o = rem >> 7;
  int p = rem & 127;
  const float* W = d ? W1 : W0;
  v2f v;
  v.x = W[o * 256 + 2 * p];
  v.y = W[o * 256 + 2 * p + 1];
  WA[(size_t)d * 65536 + (size_t)p * 512 + o] = v;
}

// ---------------------------------------------------------------------------
// k_ssm: one block per channel h (grid=256), 256 threads = 8 waves.
// Sequential over 32 chunks: load u tile (L2-resident x, with prefetch of the
// next chunk), fused Q-GEMM + Toeplitz-GEMM sharing B tiles, boundary GEMM on
// the state columns, D-skip + exact GELU, then the tiny per-chunk state chain.
// ---------------------------------------------------------------------------
__global__ __launch_bounds__(256) void k_ssm(
    const float* __restrict__ x, const float* __restrict__ Dskip,
    const v2f* __restrict__ AP, const v2f* __restrict__ QP,
    const v2f* __restrict__ WT, float* __restrict__ g, int dir) {
  __shared__ float uT[64 * 66];   // [b][t], padded rows (bank-conflict free b64)
  __shared__ v2f   sp[64 * 33];   // states [b][n], (re,im) pairs, padded
  __shared__ float rt[64 * 66];   // R tile [r][b]
  __shared__ v2f   wTs[NN];

  const int h = blockIdx.x;
  const int tid = threadIdx.x;
  const int lane = tid & 31, wv = tid >> 5;
  const int lrow = lane & 15, khalf = lane >> 4, koff = khalf * 2;

  if (tid < NN) wTs[tid] = WT[(size_t)(dir * HH + h) * NN + tid];
  for (int i = tid; i < 64 * 33; i += 256) { v2f z; z.x = 0.f; z.y = 0.f; sp[i] = z; }
  const float Dh = Dskip[h];
  const size_t apb = (size_t)(dir * HH + h) * (64 * 64);
  const size_t qpb = (size_t)(dir * HH + h) * (32 * 64);
  __syncthreads();

  const int mt = wv >> 1;            // output row tile (4 of them)
  const int nt0 = (wv & 1) * 2;      // first of two column tiles
  const int col0 = nt0 * 16 + lrow, col1 = col0 + 16;

  for (int c = 0; c < NCHUNK; ++c) {
    // 1) cooperative load of the u tile (scattered 4B reads; x lives in the
    //    192MB L2). Prefetch next chunk's lines to hide L2 latency.
    for (int i = tid; i < 64 * TT; i += 256) {
      int b = i >> 6, t = i & 63;
      int pos = dir ? (LL - 1 - (c * TT + t)) : (c * TT + t);
      uT[b * 66 + t] = x[((size_t)b * LL + pos) * HH + h];
      if (c + 1 < NCHUNK) {
        int pos2 = dir ? (LL - 1 - ((c + 1) * TT + t)) : ((c + 1) * TT + t);
        __builtin_prefetch(&x[((size_t)b * LL + pos2) * HH + h], 0, 3);
      }
    }
    __syncthreads();

    // 2) fused Q-GEMM + Toeplitz-GEMM: both consume the same B tiles.
    //    R[64 x 64b] = Q_h @ u   and   Y += M_h @ u
    v8f r0 = {}; v8f r1 = {};
    v8f y0 = {}; v8f y1 = {};
    #pragma unroll 4
    for (int ks = 0; ks < 16; ++ks) {
      int k = ks * 4;
      v2f aQ = QP[qpb + (size_t)(ks * 2 + khalf) * TT + mt * 16 + lrow];
      v2f aM = AP[apb + (size_t)(ks * 2 + khalf) * TT + mt * 16 + lrow];
      v2f b0 = *(const v2f*)&uT[col0 * 66 + k + koff];
      v2f b1 = *(const v2f*)&uT[col1 * 66 + k + koff];
      r0 = wmma4(aQ, b0, r0);
      r1 = wmma4(aQ, b1, r1);
      y0 = wmma4(aM, b0, y0);
      y1 = wmma4(aM, b1, y1);
    }
    #pragma unroll
    for (int vr = 0; vr < 8; ++vr) {
      int rr = mt * 16 + vr + khalf * 8;
      rt[rr * 66 + col0] = r0[vr];
      rt[rr * 66 + col1] = r1[vr];
    }

    // 3) boundary GEMM: Y += P_h @ s_prev (state columns, k = 64..127)
    #pragma unroll 4
    for (int ks = 16; ks < 32; ++ks) {
      int k = ks * 4;
      int n = (k + koff - 64) >> 1;    // (re,im) pair index, k+koff even
      v2f a  = AP[apb + (size_t)(ks * 2 + khalf) * TT + mt * 16 + lrow];
      v2f b0 = sp[col0 * 33 + n];
      v2f b1 = sp[col1 * 33 + n];
      y0 = wmma4(a, b0, y0);
      y1 = wmma4(a, b1, y1);
    }

    // 4) epilogue: D-skip + exact GELU, store into g[h][b*L + pos]
    //    (per-lane consecutive pos -> stores coalesce/vectorize)
    float* grow = g + (size_t)h * BL;
    #pragma unroll
    for (int vr = 0; vr < 8; ++vr) {
      int j = mt * 16 + vr + khalf * 8;
      int pos = dir ? (LL - 1 - (c * TT + j)) : (c * TT + j);
      float u0 = uT[col0 * 66 + j];
      float u1 = uT[col1 * 66 + j];
      float yv0 = y0[vr] + Dh * u0;
      float yv1 = y1[vr] + Dh * u1;
      float g0 = 0.5f * yv0 * (1.0f + erff(yv0 * 0.70710678118f));
      float g1 = 0.5f * yv1 * (1.0f + erff(yv1 * 0.70710678118f));
      grow[(size_t)col0 * LL + pos] = g0;
      grow[(size_t)col1 * LL + pos] = g1;
    }
    __syncthreads();

    // 5) state chain: s <- w^T o s + r   (tiny, once per chunk)
    for (int i = tid; i < NB * NN; i += 256) {
      int b = i >> 5, n = i & 31;
      float rr = rt[(2 * n) * 66 + b];
      float ri = rt[(2 * n + 1) * 66 + b];
      v2f s = sp[b * 33 + n];
      v2f w = wTs[n];
      v2f ns;
      ns.x = w.x * s.x - w.y * s.y + rr;
      ns.y = w.x * s.y + w.y * s.x + ri;
      sp[b * 33 + n] = ns;
    }
    __syncthreads();
  }
}

// ---------------------------------------------------------------------------
// k_glu: z[512 x 64cols] = W @ g + b, out = z1*sigmoid(z2). 512 thr = 16 waves,
// wave w owns z-row pair (16w..16w+15, 256+16w..). dir0 stores, dir1 adds.
// B-operand loads from g[h][pos] are coalesced across lanes (contiguous pos).
// ---------------------------------------------------------------------------
__global__ __launch_bounds__(512) void k_glu(
    const float* __restrict__ g, const v2f* __restrict__ WA,
    const float* __restrict__ bglu, float* __restrict__ out, int add) {
  const int tid = threadIdx.x;
  const int lane = tid & 31, wv = tid >> 5;     // wv = 0..15
  const int lrow = lane & 15, khalf = lane >> 4, koff = khalf * 2;
  const size_t P0 = (size_t)blockIdx.x * 64;

  v8f accT[4] = {{}, {}, {}, {}};
  v8f accB[4] = {{}, {}, {}, {}};

  for (int ks = 0; ks < 64; ++ks) {
    int k = ks * 4;
    size_t prow = (size_t)(ks * 2 + khalf) * 512;
    v2f aT = WA[prow + wv * 16 + lrow];
    v2f aB = WA[prow + 256 + wv * 16 + lrow];
    const float* gk0 = g + (size_t)(k + koff) * BL + P0 + lrow;
    const float* gk1 = gk0 + BL;
    #pragma unroll
    for (int nt = 0; nt < 4; ++nt) {
      v2f bb;
      bb.x = gk0[nt * 16];
      bb.y = gk1[nt * 16];
      accT[nt] = wmma4(aT, bb, accT[nt]);
      accB[nt] = wmma4(aB, bb, accB[nt]);
    }
  }
  #pragma unroll
  for (int nt = 0; nt < 4; ++nt) {
    #pragma unroll
    for (int vr = 0; vr < 8; ++vr) {
      int o1 = wv * 16 + vr + khalf * 8;
      size_t p = P0 + nt * 16 + lrow;
      float z1 = accT[nt][vr] + bglu[o1];
      float z2 = accB[nt][vr] + bglu[o1 + 256];
      float v = z1 * (1.0f / (1.0f + expf(-z2)));
      float* dst = out + p * HH + o1;
      if (add) *dst = *dst + v; else *dst = v;
    }
  }
}

// ---------------------------------------------------------------------------
extern "C" void kernel_launch(void* const* d_in, const int* in_sizes, int n_in,
                              void* d_out, int out_size, void* d_ws, size_t ws_size,
                              hipStream_t stream) {
  (void)in_sizes; (void)n_in; (void)out_size; (void)ws_size;
  const float* x    = (const float*)d_in[0];
  const float* ldF  = (const float*)d_in[1];
  const float* crF  = (const float*)d_in[2];
  const float* ciF  = (const float*)d_in[3];
  const float* larF = (const float*)d_in[4];
  const float* aimF = (const float*)d_in[5];
  const float* dskF = (const float*)d_in[6];
  const float* WF   = (const float*)d_in[7];
  const float* bF   = (const float*)d_in[8];
  const float* ldR  = (const float*)d_in[9];
  const float* crR  = (const float*)d_in[10];
  const float* ciR  = (const float*)d_in[11];
  const float* larR = (const float*)d_in[12];
  const float* aimR = (const float*)d_in[13];
  const float* dskR = (const float*)d_in[14];
  const float* WR   = (const float*)d_in[15];
  const float* bR   = (const float*)d_in[16];
  float* out = (float*)d_out;

  char* ws = (char*)d_ws;
  size_t off = 0;
  v2f* WT = (v2f*)(ws + off); off += (size_t)2 * HH * NN * 8;          // 128 KB
  v2f* AP = (v2f*)(ws + off); off += (size_t)2 * HH * 64 * 64 * 8;    // 16 MB
  v2f* QP = (v2f*)(ws + off); off += (size_t)2 * HH * 32 * 64 * 8;    // 8 MB
  v2f* WA = (v2f*)(ws + off); off += (size_t)2 * 128 * 512 * 8;       // 1 MB
  float* G = (float*)(ws + off); off += (size_t)HH * BL * 4;          // 134 MB

  k_prep<<<512, 64, 0, stream>>>(ldF, crF, ciF, larF, aimF,
                                 ldR, crR, ciR, larR, aimR, WT, AP, QP);
  k_packW<<<512, 256, 0, stream>>>(WF, WR, WA);

  // forward direction
  k_ssm<<<HH, 256, 0, stream>>>(x, dskF, AP, QP, WT, G, 0);
  k_glu<<<BL / 64, 512, 0, stream>>>(G, WA, bF, out, 0);
  // reverse direction (accumulates; ordered on stream, no atomics needed)
  k_ssm<<<HH, 256, 0, stream>>>(x, dskR, AP, QP, WT, G, 1);
  k_glu<<<BL / 64, 512, 0, stream>>>(G, WA + 65536, bR, out, 1);
}